// LSTMEncoder_78975858639682
// MI455X (gfx1250) — compile-verified
//
#include <hip/hip_runtime.h>
#include <hip/hip_bf16.h>

// ---------------------------------------------------------------------------
// Fused 2-layer LSTM (H=64) + FC + tanh for MI455X (gfx1250), fp32 WMMA path.
//   grid  : 128 workgroups, one per 16-row batch tile (B=2048)
//   block : 512 threads = 16 wave32; wave w owns gate columns [16w, 16w+16)
//   WMMA  : V_WMMA_F32_16X16X4_F32  (D = A(16x4) * B(4x16) + C, fp32)
// Weights (B operands) preloaded to VGPRs; h/c state stays in LDS/registers;
// both layers fused per timestep so h1_seq is never materialized.
// Latency optimizations for the sequential recurrence:
//   - x[t+1] prefetched into a register across iterations (no loadcnt wait
//     in front of the WMMA chain)
//   - each GEMM accumulated in TWO independent WMMA chains (halves the
//     serialized D->C dependency depth), merged with v_pk_add_f32
//   - activations via native v_tanh_f32 / v_exp_f32 / v_rcp_f32
// ---------------------------------------------------------------------------

typedef __attribute__((ext_vector_type(2))) float v2f;
typedef __attribute__((ext_vector_type(8))) float v8f;

#define LSTM_B    2048
#define LSTM_T    512
#define LSTM_H    64
#define LSTM_G    256          // 4*H
#define MTILE     16           // batch rows per workgroup
#define NWAVES    16           // one wave per 16-wide gate N-tile
#define THREADS   (NWAVES * 32)
#define GPAD      260          // gate-row stride in floats (bank-conflict pad)
#define HPAD      68           // h-row stride in floats   (bank-conflict pad)

#define LOG2E_F   1.4426950408889634f

__device__ __forceinline__ float fast_tanh(float x) {
#if __has_builtin(__builtin_amdgcn_tanhf)
    return __builtin_amdgcn_tanhf(x);          // v_tanh_f32
#else
    float r;
    // TRANS-op hazard: 1 independent op (v_nop) before result use.
    asm volatile("v_tanh_f32 %0, %1\n\tv_nop" : "=v"(r) : "v"(x));
    return r;
#endif
}

__device__ __forceinline__ float fast_sigmoid(float x) {
    // sigmoid(x) = 1 / (1 + 2^(-x*log2(e)))  ->  v_exp_f32 + v_add + v_rcp_f32
    float e = __builtin_amdgcn_exp2f(x * -LOG2E_F);
    return __builtin_amdgcn_rcpf(1.0f + e);
}

__device__ __forceinline__ v8f wmma_f32_k4(v2f a, v2f b, v8f c) {
    // 8 args: (neg_a, A, neg_b, B, c_mod, C, reuse_a, reuse_b)
    return __builtin_amdgcn_wmma_f32_16x16x4_f32(false, a, false, b,
                                                 (short)0, c, false, false);
}

__global__ __launch_bounds__(THREADS, 1)
void LSTMEncoder_78975858639682_kernel(
    const float* __restrict__ x,      // (2048, 512, 4)
    const float* __restrict__ w_ih0,  // (256, 4)
    const float* __restrict__ w_hh0,  // (256, 64)
    const float* __restrict__ b_ih0,  // (256)
    const float* __restrict__ b_hh0,  // (256)
    const float* __restrict__ w_ih1,  // (256, 64)
    const float* __restrict__ w_hh1,  // (256, 64)
    const float* __restrict__ b_ih1,  // (256)
    const float* __restrict__ b_hh1,  // (256)
    const float* __restrict__ fc_w,   // (64, 64)
    const float* __restrict__ fc_b,   // (64)
    float* __restrict__ out)          // (2048, 64)
{
    __shared__ float sG [MTILE * GPAD];   // gate matrix 16 x 256 (padded)
    __shared__ float sH0[MTILE * HPAD];   // layer-0 hidden 16 x 64 (padded)
    __shared__ float sH1[MTILE * HPAD];   // layer-1 hidden 16 x 64 (padded)

    const int tid  = threadIdx.x;
    const int lane = tid & 31;
    const int wave = tid >> 5;
    const int half = lane >> 4;      // 0: lanes 0-15, 1: lanes 16-31
    const int l16  = lane & 15;
    const int rowBase = blockIdx.x * MTILE;

    // ---- preload B-operand weight tiles into VGPRs (loop invariant) ----
    // B(4x16) per-lane layout: N = lane%16, VGPR0/1 hold K = 2*half, 2*half+1.
    // B[k][n] = W^T[k][n] = W[n][k]  -> contiguous float2 in row-major W.
    const int ncol = wave * 16 + l16;          // this lane's gate column
    const int kb   = 2 * half;                 // K pair base within chunk
    v2f wih0 = *(const v2f*)(w_ih0 + ncol * 4 + kb);
    v2f whh0[16], wih1[16], whh1[16];
#pragma unroll
    for (int kc = 0; kc < 16; ++kc) {
        const int k = kc * 4 + kb;
        whh0[kc] = *(const v2f*)(w_hh0 + ncol * LSTM_H + k);
        wih1[kc] = *(const v2f*)(w_ih1 + ncol * LSTM_H + k);
        whh1[kc] = *(const v2f*)(w_hh1 + ncol * LSTM_H + k);
    }
    const float bias0 = b_ih0[ncol] + b_hh0[ncol];
    const float bias1 = b_ih1[ncol] + b_hh1[ncol];

    // ---- element-wise mapping: thread -> (row = wave, cols = 2*lane,2*lane+1)
    const int   erow = wave;
    const int   ecol = lane * 2;
    float c0x = 0.f, c0y = 0.f, c1x = 0.f, c1y = 0.f;   // cell state in regs

    // zero-init hidden state buffers
    for (int i = tid; i < MTILE * HPAD; i += THREADS) { sH0[i] = 0.f; sH1[i] = 0.f; }
    __syncthreads();

    const int arow = l16;                      // A-tile row for this lane
    const float* xrow = x + (size_t)(rowBase + arow) * LSTM_T * 4 + kb;

    // software-pipelined x operand: load t=0 up front
    v2f ax = *(const v2f*)(xrow);

#pragma unroll 1
    for (int t = 0; t < LSTM_T; ++t) {
        // prefetch next step's x A-tile (clamped at the tail) -- its latency
        // overlaps the whole step body.
        const int tn = (t < LSTM_T - 1) ? (t + 1) : t;
        v2f ax_next = *(const v2f*)(xrow + (size_t)tn * 4);

        // ========== layer 0: gates = b + x_t@Wih0^T + h0@Whh0^T =============
        // two independent accumulation chains to halve D->C serialization
        v8f accA, accB;
#pragma unroll
        for (int j = 0; j < 8; ++j) { accA[j] = bias0; accB[j] = 0.f; }
        accB = wmma_f32_k4(ax, wih0, accB);
#pragma unroll
        for (int kc = 0; kc < 16; kc += 2) {
            v2f ahA = *(const v2f*)(sH0 + arow * HPAD + (kc    ) * 4 + kb);
            v2f ahB = *(const v2f*)(sH0 + arow * HPAD + (kc + 1) * 4 + kb);
            accA = wmma_f32_k4(ahA, whh0[kc],     accA);
            accB = wmma_f32_k4(ahB, whh0[kc + 1], accB);
        }
        // merge chains + store D (VGPR j -> row (j + 8*half), col = ncol)
#pragma unroll
        for (int j = 0; j < 8; ++j)
            sG[(j + 8 * half) * GPAD + ncol] = accA[j] + accB[j];
        __syncthreads();

        // ========== layer 0 element-wise ====================================
        {
            v2f gi = *(const v2f*)(sG + erow * GPAD + ecol);
            v2f gf = *(const v2f*)(sG + erow * GPAD + 64  + ecol);
            v2f gg = *(const v2f*)(sG + erow * GPAD + 128 + ecol);
            v2f go = *(const v2f*)(sG + erow * GPAD + 192 + ecol);
            float i0 = fast_sigmoid(gi.x), i1 = fast_sigmoid(gi.y);
            float f0 = fast_sigmoid(gf.x), f1 = fast_sigmoid(gf.y);
            float g0 = fast_tanh(gg.x),    g1 = fast_tanh(gg.y);
            float o0 = fast_sigmoid(go.x), o1 = fast_sigmoid(go.y);
            c0x = f0 * c0x + i0 * g0;
            c0y = f1 * c0y + i1 * g1;
            v2f hn; hn.x = o0 * fast_tanh(c0x); hn.y = o1 * fast_tanh(c0y);
            *(v2f*)(sH0 + erow * HPAD + ecol) = hn;
        }
        __syncthreads();

        // ========== layer 1: gates = b + h0@Wih1^T + h1@Whh1^T ==============
        // natural dual-chain split: chain A over Wih1/sH0, chain B over Whh1/sH1
#pragma unroll
        for (int j = 0; j < 8; ++j) { accA[j] = bias1; accB[j] = 0.f; }
#pragma unroll
        for (int kc = 0; kc < 16; ++kc) {
            v2f ahA = *(const v2f*)(sH0 + arow * HPAD + kc * 4 + kb);
            v2f ahB = *(const v2f*)(sH1 + arow * HPAD + kc * 4 + kb);
            accA = wmma_f32_k4(ahA, wih1[kc], accA);
            accB = wmma_f32_k4(ahB, whh1[kc], accB);
        }
#pragma unroll
        for (int j = 0; j < 8; ++j)
            sG[(j + 8 * half) * GPAD + ncol] = accA[j] + accB[j];
        __syncthreads();

        // ========== layer 1 element-wise ====================================
        {
            v2f gi = *(const v2f*)(sG + erow * GPAD + ecol);
            v2f gf = *(const v2f*)(sG + erow * GPAD + 64  + ecol);
            v2f gg = *(const v2f*)(sG + erow * GPAD + 128 + ecol);
            v2f go = *(const v2f*)(sG + erow * GPAD + 192 + ecol);
            float i0 = fast_sigmoid(gi.x), i1 = fast_sigmoid(gi.y);
            float f0 = fast_sigmoid(gf.x), f1 = fast_sigmoid(gf.y);
            float g0 = fast_tanh(gg.x),    g1 = fast_tanh(gg.y);
            float o0 = fast_sigmoid(go.x), o1 = fast_sigmoid(go.y);
            c1x = f0 * c1x + i0 * g0;
            c1y = f1 * c1y + i1 * g1;
            v2f hn; hn.x = o0 * fast_tanh(c1x); hn.y = o1 * fast_tanh(c1y);
            *(v2f*)(sH1 + erow * HPAD + ecol) = hn;
        }
        __syncthreads();

        ax = ax_next;
    }

    // ================= FC epilogue: out = tanh(h1 @ fc_w^T + fc_b) =========
    if (wave < 4) {
        const int nco = wave * 16 + l16;   // output column
        v8f accA, accB;
        const float fb = fc_b[nco];
#pragma unroll
        for (int j = 0; j < 8; ++j) { accA[j] = fb; accB[j] = 0.f; }
#pragma unroll
        for (int kc = 0; kc < 16; kc += 2) {
            v2f ahA = *(const v2f*)(sH1 + arow * HPAD + (kc    ) * 4 + kb);
            v2f ahB = *(const v2f*)(sH1 + arow * HPAD + (kc + 1) * 4 + kb);
            v2f bwA = *(const v2f*)(fc_w + nco * LSTM_H + (kc    ) * 4 + kb);
            v2f bwB = *(const v2f*)(fc_w + nco * LSTM_H + (kc + 1) * 4 + kb);
            accA = wmma_f32_k4(ahA, bwA, accA);
            accB = wmma_f32_k4(ahB, bwB, accB);
        }
#pragma unroll
        for (int j = 0; j < 8; ++j) {
            const int r = j + 8 * half;
            out[(size_t)(rowBase + r) * LSTM_H + nco] = fast_tanh(accA[j] + accB[j]);
        }
    }
}

extern "C" void kernel_launch(void* const* d_in, const int* in_sizes, int n_in,
                              void* d_out, int out_size, void* d_ws, size_t ws_size,
                              hipStream_t stream) {
    (void)in_sizes; (void)n_in; (void)d_ws; (void)ws_size; (void)out_size;
    const float* x     = (const float*)d_in[0];
    const float* w_ih0 = (const float*)d_in[1];
    const float* w_hh0 = (const float*)d_in[2];
    const float* b_ih0 = (const float*)d_in[3];
    const float* b_hh0 = (const float*)d_in[4];
    const float* w_ih1 = (const float*)d_in[5];
    const float* w_hh1 = (const float*)d_in[6];
    const float* b_ih1 = (const float*)d_in[7];
    const float* b_hh1 = (const float*)d_in[8];
    const float* fc_w  = (const float*)d_in[9];
    const float* fc_b  = (const float*)d_in[10];
    float* out = (float*)d_out;

    dim3 grid(LSTM_B / MTILE);   // 128 workgroups
    dim3 block(THREADS);         // 512 threads = 16 wave32
    LSTMEncoder_78975858639682_kernel<<<grid, block, 0, stream>>>(
        x, w_ih0, w_hh0, b_ih0, b_hh0, w_ih1, w_hh1, b_ih1, b_hh1, fc_w, fc_b, out);
}